// RandomElementFI_8976481649188
// MI455X (gfx1250) — compile-verified
//
#include <hip/hip_runtime.h>

// RandomElementFI on MI455X (gfx1250):
//   out = copy(x)  (256 MB fp32, pure streaming)  then  out[idx[i]] = vals[i]
// Bandwidth-bound: ~520 MB total -> ~23 us at 23.3 TB/s. No FLOPs, so the
// CDNA5 feature that matters is the async Global<->LDS DMA path (ASYNCcnt),
// b128 per-lane transfers, and NT temporal hints on the dead input stream.

#define TPB   256
#define DEPTH 8   // float4 per thread per tile stage -> 32 KB LDS tile / block

__global__ __launch_bounds__(TPB) void fi_copy_async(const float* __restrict__ src,
                                                     float* __restrict__ dst,
                                                     long long n) {
    __shared__ float4 stage[TPB * DEPTH];  // 32 KB of the 320 KB WGP LDS
    const unsigned tid = threadIdx.x;
    // Low 32 bits of the generic address of a __shared__ object are the
    // workgroup-relative LDS byte offset (flat = {shared_aperture, lds_off}).
    const unsigned ldsBase = (unsigned)(size_t)&stage[0];

    const long long n4        = n >> 2;                  // float4 count
    const long long tileQuads = (long long)TPB * DEPTH;  // 2048 float4 / tile
    const long long tiles     = n4 / tileQuads;          // 8192 for this shape

    for (long long t = blockIdx.x; t < tiles; t += gridDim.x) {
        // Byte offset of tile start; max (2^28 - 32KB) so it fits i32 VADDR.
        const unsigned byteBase = (unsigned)(t * tileQuads * 16ll);

        // Stage 1: issue DEPTH async 128-bit DMA loads HBM -> LDS.
        // One wave32 instruction moves 512 B; a block posts a 32 KB tile.
        // th:TH_LOAD_NT — input is read exactly once, keep it out of L2.
#pragma unroll
        for (int k = 0; k < DEPTH; ++k) {
            unsigned goff = byteBase + (k * TPB + tid) * 16u;
            unsigned loff = ldsBase  + (k * TPB + tid) * 16u;
            asm volatile("global_load_async_to_lds_b128 %0, %1, %2 th:TH_LOAD_NT"
                         :: "v"(loff), "v"(goff), "s"(src) : "memory");
        }
        // Wave-private dependency: each lane stores only bytes it loaded,
        // so a per-wave asynccnt drain is sufficient (no workgroup barrier).
        asm volatile("s_wait_asynccnt 0x0" ::: "memory");

        // Stage 2: issue DEPTH async 128-bit DMA stores LDS -> HBM.
        // Default RT (wr-rinse) store: data is pushed to HBM but lines may
        // stay resident, letting the follow-up scatter hit in the 192 MB L2.
#pragma unroll
        for (int k = 0; k < DEPTH; ++k) {
            unsigned goff = byteBase + (k * TPB + tid) * 16u;
            unsigned loff = ldsBase  + (k * TPB + tid) * 16u;
            asm volatile("global_store_async_from_lds_b128 %0, %1, %2"
                         :: "v"(goff), "v"(loff), "s"(dst) : "memory");
        }
        // Drain before the next iteration reuses the LDS staging slots.
        asm volatile("s_wait_asynccnt 0x0" ::: "memory");
    }

    // Generic remainder paths (empty for the fixed 64x1024x1024 shape).
    const long long gtid     = (long long)blockIdx.x * TPB + tid;
    const long long gthreads = (long long)gridDim.x * TPB;
    const float4* s4 = (const float4*)src;
    float4*       d4 = (float4*)dst;
    for (long long i = tiles * tileQuads + gtid; i < n4; i += gthreads)
        d4[i] = s4[i];
    for (long long i = (n4 << 2) + gtid; i < n; i += gthreads)
        dst[i] = src[i];
}

__global__ __launch_bounds__(TPB) void fi_scatter(float* __restrict__ dst,
                                                  const float* __restrict__ vals,
                                                  const int* __restrict__ idx,
                                                  int m) {
    int i = blockIdx.x * TPB + threadIdx.x;
    if (i < m) dst[idx[i]] = vals[i];   // unique indices -> no atomics needed
}

extern "C" void kernel_launch(void* const* d_in, const int* in_sizes, int n_in,
                              void* d_out, int out_size, void* d_ws, size_t ws_size,
                              hipStream_t stream) {
    (void)n_in; (void)d_ws; (void)ws_size; (void)out_size;
    const float* x    = (const float*)d_in[0];   // x: 64*1024*1024 fp32
    const float* vals = (const float*)d_in[1];   // fault_vals: 671089 fp32
    const int*   idx  = (const int*)d_in[2];     // fault_idx:  671089 int32
    float* out = (float*)d_out;

    const long long n = in_sizes[0];
    const int       m = in_sizes[1];

    // 2048 blocks x 8 waves: 8192 tiles -> 4 tiles/block; far more waves than
    // WGPs, so async-DMA latency is hidden by wave-level parallelism.
    fi_copy_async<<<2048, TPB, 0, stream>>>(x, out, n);
    fi_scatter<<<(m + TPB - 1) / TPB, TPB, 0, stream>>>(out, vals, idx, m);
}